// MorsePotentialSpecies_63101659513297
// MI455X (gfx1250) — compile-verified
//
#include <hip/hip_runtime.h>
#include <math.h>

// ---------------- problem constants (from reference) ----------------
#define N_PART 8192
#define NCT    4
#define NT     (N_PART / 16)     // 512 16-wide tiles per dimension
#define CHUNK  16                // column tiles swept per wave
#define NCHUNK (NT / CHUNK)      // 32 chunks per tile-row
#define WAVES_PER_BLOCK 8
#define BLOCK  256

#define EPS_MAX   5.0f
#define EPS_MIN   1.0f
#define ALPHA_MAX 3.0f
#define ALPHA_MIN 1.0f
#define RO2       (1.7f * 1.7f)  // r_onset^2
#define RC2       (2.0f * 2.0f)  // r_cutoff^2
// 1 / (r_c^2 - r_o^2)^3
#define SMOOTH_INV (1.0f / ((RC2 - RO2) * (RC2 - RO2) * (RC2 - RO2)))

typedef __attribute__((ext_vector_type(2))) float v2f;
typedef __attribute__((ext_vector_type(8))) float v8f;

// ---------------- prep: per-particle packing ----------------
// pq[i] = (x, y, z, |p|^2) ; rc[i] = (radius, bitcast(species))
__global__ void prep_particles(const float* __restrict__ celltype,
                               const float* __restrict__ radius,
                               const float* __restrict__ pos,
                               float4* __restrict__ pq,
                               float2* __restrict__ rc) {
    int i = blockIdx.x * blockDim.x + threadIdx.x;
    if (i >= N_PART) return;
    float x = pos[3 * i + 0];
    float y = pos[3 * i + 1];
    float z = pos[3 * i + 2];
    pq[i] = make_float4(x, y, z, x * x + y * y + z * z);
    // one-hot -> species index
    const float* c = celltype + 4 * i;
    int ct = 0;
    float best = c[0];
#pragma unroll
    for (int k = 1; k < NCT; ++k) {
        if (c[k] > best) { best = c[k]; ct = k; }
    }
    rc[i] = make_float2(radius[i], __int_as_float(ct));
}

// ---------------- prep: 4x4 species LUTs + zero the output ----------------
__global__ void prep_tables(const float* __restrict__ eps_raw,
                            const float* __restrict__ alpha_raw,
                            float* __restrict__ eps_tab,
                            float* __restrict__ alpha_tab,
                            float* __restrict__ out) {
    int t = threadIdx.x;
    if (t == 0) out[0] = 0.0f;   // harness does not re-zero d_out between replays
    if (t < 16) {
        int i = t >> 2, j = t & 3;
        float me = (i == j) ? eps_raw[i * NCT + i]
                            : 0.5f * (eps_raw[i * NCT + j] + eps_raw[j * NCT + i]);
        float ma = (i == j) ? alpha_raw[i * NCT + i]
                            : 0.5f * (alpha_raw[i * NCT + j] + alpha_raw[j * NCT + i]);
        eps_tab[t]   = (1.0f / (1.0f + expf(-me))) * EPS_MAX   + EPS_MIN;
        alpha_tab[t] = (1.0f / (1.0f + expf(-ma))) * ALPHA_MAX + ALPHA_MIN;
    }
}

// ---------------- main: WMMA-tiled pairwise Morse energy ----------------
// One wave32 owns row-tile ti (16 particles) and sweeps CHUNK column tiles.
// V_WMMA_F32_16X16X4_F32 computes D = (-2*P_rows) x (P_cols)^T + (qm+qn) = dr^2
// directly in the tensor pipe (A scaled by -2, |p|^2 sums fed via C).
__global__ void __launch_bounds__(BLOCK)
morse_tiles(const float4* __restrict__ pq,
            const float2* __restrict__ rc,
            const float*  __restrict__ eps_tab,
            const float*  __restrict__ alpha_tab,
            float* __restrict__ out) {
    __shared__ float s_eps[16];
    __shared__ float s_alpha[16];
    int tid = threadIdx.x;
    if (tid < 16)       s_eps[tid]        = eps_tab[tid];
    else if (tid < 32)  s_alpha[tid - 16] = alpha_tab[tid - 16];
    __syncthreads();

    const int lane  = tid & 31;
    const int hb    = lane >> 4;        // half-wave: 0 -> K=0,1 ; 1 -> K=2,3
    const int ln    = lane & 15;
    const int w     = blockIdx.x * WAVES_PER_BLOCK + (tid >> 5);
    const int ti    = w / NCHUNK;       // row tile [0, 512)
    const int chunk = w % NCHUNK;       // column chunk [0, 32)

    // A fragment: 16x4 f32 layout (ISA 7.12.2): lanes 0-15 hold K=0/1,
    // lanes 16-31 hold K=2/3, M = lane&15. Scale by -2 so D = -2*dot + C.
    const float4 pm = pq[ti * 16 + ln];
    v2f a;
    a[0] = -2.0f * (hb ? pm.z : pm.x);
    a[1] = -2.0f * (hb ? 0.0f : pm.y);

    // Row-side scalars for this lane's 8 output rows (C/D layout:
    // VGPR r, lanes 0-15 -> M=r, lanes 16-31 -> M=r+8).
    const int rowbase = ti * 16 + hb * 8;
    float qm[8], radm[8];
    int   ctm[8];
#pragma unroll
    for (int r = 0; r < 8; ++r) {
        const int m = rowbase + r;
        qm[r] = pq[m].w;
        float2 v = rc[m];
        radm[r] = v.x;
        ctm[r]  = __float_as_int(v.y);
    }

    // This lane's diagonal element sits at row slot rdiag (may be out of
    // [0,8) -> never matches, i.e. the diagonal lives in the other half-wave).
    const int rdiag = ln - hb * 8;

    float acc = 0.0f;
    for (int jc = 0; jc < CHUNK; ++jc) {
        const int tj = chunk * CHUNK + jc;
        const int n  = tj * 16 + ln;
        const float4 pn  = pq[n];
        const float2 rcn = rc[n];
        const int    ctn = __float_as_int(rcn.y);

        // B fragment: 4x16 f32, same striping as A (row K across half-waves).
        v2f b;
        b[0] = hb ? pn.z : pn.x;
        b[1] = hb ? 0.0f : pn.y;

        // C = |p_m|^2 + |p_n|^2  (per C/D VGPR layout)
        v8f c;
#pragma unroll
        for (int r = 0; r < 8; ++r) c[r] = qm[r] + pn.w;

        // D[r] = dr^2 for pair (rowbase+r, n)
        v8f d = __builtin_amdgcn_wmma_f32_16x16x4_f32(
            /*neg_a=*/false, a, /*neg_b=*/false, b,
            /*c_mod=*/(short)0, c, /*reuse_a=*/false, /*reuse_b=*/false);

        // Single cheap any-in-range test over the lane's 8 pairs (min tree).
        // The diagonal (dr^2 == 0) drags its one tile per wave into the
        // guarded block; it's rejected per-element below.
        float dmin = fminf(fminf(fminf(d[0], d[1]), fminf(d[2], d[3])),
                           fminf(fminf(d[4], d[5]), fminf(d[6], d[7])));

        // ~98% of tiles have no lane passing -> whole tail skipped via execz
        if (dmin < RC2) {
            const bool same_tile = (ti == tj);
#pragma unroll
            for (int r = 0; r < 8; ++r) {
                float dr2 = d[r];
                const bool ok = (dr2 < RC2) && !(same_tile && (r == rdiag));
                if (ok) {
                    dr2 = fmaxf(dr2, 0.0f);
                    const float s = (dr2 < RO2)
                        ? 1.0f
                        : (RC2 - dr2) * (RC2 - dr2) *
                          (RC2 + 2.0f * dr2 - 3.0f * RO2) * SMOOTH_INV;
                    // dr2 in [0,4]: raw v_sqrt_f32 is safe
                    const float dr    = __builtin_amdgcn_sqrtf(dr2);
                    const int   idx   = ctm[r] * NCT + ctn;
                    const float eps   = s_eps[idx];
                    const float alp   = s_alpha[idx];
                    const float sigma = radm[r] + rcn.x;
                    // exp arg in ~[-6, 4]: fast v_exp_f32 path is safe
                    const float t     = 1.0f - __expf(-alp * (dr - sigma));
                    acc += s * (eps * t * t - eps);
                }
            }
        }
    }

    // wave32 reduction, one atomic per wave
    for (int off = 16; off > 0; off >>= 1)
        acc += __shfl_down(acc, off, 32);
    if (lane == 0) atomicAdd(out, 0.5f * acc);
}

// ---------------- entry point ----------------
extern "C" void kernel_launch(void* const* d_in, const int* in_sizes, int n_in,
                              void* d_out, int out_size, void* d_ws, size_t ws_size,
                              hipStream_t stream) {
    const float* eps_raw   = (const float*)d_in[0];   // [4,4]
    const float* alpha_raw = (const float*)d_in[1];   // [4,4]
    const float* celltype  = (const float*)d_in[2];   // [N,4] one-hot
    const float* radius    = (const float*)d_in[3];   // [N,1]
    const float* positions = (const float*)d_in[4];   // [N,3]
    float* out = (float*)d_out;

    // workspace layout: pq (128KB) | rc (64KB) | eps_tab (64B) | alpha_tab (64B)
    char* ws = (char*)d_ws;
    float4* pq        = (float4*)ws;
    float2* rc        = (float2*)(ws + (size_t)N_PART * sizeof(float4));
    float*  eps_tab   = (float*)(ws + (size_t)N_PART * (sizeof(float4) + sizeof(float2)));
    float*  alpha_tab = eps_tab + 16;

    prep_particles<<<N_PART / 256, 256, 0, stream>>>(celltype, radius, positions, pq, rc);
    prep_tables<<<1, 16, 0, stream>>>(eps_raw, alpha_raw, eps_tab, alpha_tab, out);

    const int total_waves = NT * NCHUNK;                 // 16384
    morse_tiles<<<total_waves / WAVES_PER_BLOCK, BLOCK, 0, stream>>>(
        pq, rc, eps_tab, alpha_tab, out);
}